// ContrastiveLoss_67439576482171
// MI455X (gfx1250) — compile-verified
//
#include <hip/hip_runtime.h>

typedef float v2f __attribute__((ext_vector_type(2)));
typedef float v8f __attribute__((ext_vector_type(8)));

#define B_ 8
#define N_ 2048
#define D_ 128
#define EPS_ 1e-6f
#define MARGIN_ 1.0f

#define BM 128   // block tile rows (t rows)
#define BN 64    // block tile cols (t1 rows)
#define BK 32    // K panel staged in LDS
#define LDP 36   // padded LDS row stride (36*4B = 144B; conflict-free across rows)
#define NPANEL (D_ / BK)   // 4 K panels

// ---------------------------------------------------------------------------
// s_wait_asynccnt helper (async global->LDS ops are tracked by ASYNCcnt).
// ---------------------------------------------------------------------------
__device__ __forceinline__ void wait_asynccnt_6() {
#if __has_builtin(__builtin_amdgcn_s_wait_asynccnt)
  __builtin_amdgcn_s_wait_asynccnt(6);
#else
  asm volatile("s_wait_asynccnt 0x6" ::: "memory");
#endif
}
__device__ __forceinline__ void wait_asynccnt_0() {
#if __has_builtin(__builtin_amdgcn_s_wait_asynccnt)
  __builtin_amdgcn_s_wait_asynccnt(0);
#else
  asm volatile("s_wait_asynccnt 0x0" ::: "memory");
#endif
}

// GLOBAL_LOAD_ASYNC_TO_LDS_B128: DMA 16B from global to LDS, no VGPR staging.
// VDST = wave-relative LDS byte offset (low 32 bits of generic LDS pointer,
// per ISA 10.2).
__device__ __forceinline__ void async_g2l_b128(const float* gaddr, const float* laddr) {
  const unsigned lds_off = (unsigned)(uintptr_t)laddr;
  asm volatile("global_load_async_to_lds_b128 %0, %1, off"
               :: "v"(lds_off), "v"(gaddr) : "memory");
}

// ---------------------------------------------------------------------------
// Kernel 1: per-row sum(x) and sum(x^2) for both embedding tensors.
// One wave32 per row: D=128 floats == one float4 per lane.
// ---------------------------------------------------------------------------
__global__ void __launch_bounds__(256)
rowstats_kernel(const float* __restrict__ et, const float* __restrict__ et1,
                float* __restrict__ sq_t, float* __restrict__ sum_t,
                float* __restrict__ sq_t1, float* __restrict__ sum_t1) {
  const int wave = threadIdx.x >> 5;
  const int lane = threadIdx.x & 31;
  const int rowG = blockIdx.x * 8 + wave;          // 0 .. 2*B_*N_-1
  const float* src; float* dsq; float* dsum; int row;
  if (rowG < B_ * N_) { src = et;  dsq = sq_t;  dsum = sum_t;  row = rowG; }
  else                { src = et1; dsq = sq_t1; dsum = sum_t1; row = rowG - B_ * N_; }
  const float4 v = ((const float4*)(src + (size_t)row * D_))[lane];
  float s = v.x + v.y + v.z + v.w;
  float q = v.x * v.x + v.y * v.y + v.z * v.z + v.w * v.w;
#pragma unroll
  for (int off = 16; off > 0; off >>= 1) {
    s += __shfl_xor(s, off, 32);
    q += __shfl_xor(q, off, 32);
  }
  if (lane == 0) { dsq[row] = q; dsum[row] = s; }
}

// ---------------------------------------------------------------------------
// Kernel 2: fused f32 WMMA GEMM (dots = E_t . E_t1^T) + contrastive-loss
// epilogue + deterministic per-block reduction. Never materializes [B,N,N].
// Global->LDS staging uses async DMA (ASYNCcnt) with double-buffered panels.
// ---------------------------------------------------------------------------
__global__ void __launch_bounds__(256)
loss_gemm_kernel(const float* __restrict__ et, const float* __restrict__ et1,
                 const int* __restrict__ ids_t, const int* __restrict__ ids_t1,
                 const float* __restrict__ sq_t, const float* __restrict__ sum_t,
                 const float* __restrict__ sq_t1, const float* __restrict__ sum_t1,
                 float* __restrict__ partials) {
  __shared__ __align__(16) float As[2][BM * LDP];
  __shared__ __align__(16) float Bs[2][BN * LDP];
  __shared__ float sqm[BM], smm[BM], sqn[BN], smn[BN];
  __shared__ int   idm[BM], idn[BN];
  __shared__ float red[256];

  const int tid  = threadIdx.x;
  const int lane = tid & 31;
  const int wave = tid >> 5;
  const int hlf  = lane >> 4;      // 0: lanes 0-15, 1: lanes 16-31
  const int l16  = lane & 15;
  const int b    = blockIdx.z;
  const int m_blk = blockIdx.x * BM;
  const int n_blk = blockIdx.y * BN;

  const float* A  = et  + ((size_t)b * N_ + m_blk) * D_;
  const float* Bt = et1 + ((size_t)b * N_ + n_blk) * D_;

  // Per-thread staging map (6 x 16B per panel: 4 for A, 2 for B).
  const int fa_r[4]  = { (tid + 0) >> 3, (tid + 256) >> 3, (tid + 512) >> 3, (tid + 768) >> 3 };
  const int fa_c4    = (tid & 7) << 2;   // same column group for all 4 (f & 7 invariant mod 256)
  const int fb_r[2]  = { (tid + 0) >> 3, (tid + 256) >> 3 };

  // Issue async loads for one K panel into buffer `buf`.
  auto issue_panel = [&](int kc, int buf) {
#pragma unroll
    for (int i = 0; i < 4; ++i)
      async_g2l_b128(A + (size_t)fa_r[i] * D_ + kc + fa_c4,
                     &As[buf][fa_r[i] * LDP + fa_c4]);
#pragma unroll
    for (int i = 0; i < 2; ++i)
      async_g2l_b128(Bt + (size_t)fb_r[i] * D_ + kc + fa_c4,
                     &Bs[buf][fb_r[i] * LDP + fa_c4]);
  };

  // Stage row stats + IDs for this block's rows/cols (overlaps with panel 0 DMA).
  issue_panel(0, 0);
  if (tid < BM) {
    sqm[tid] = sq_t [b * N_ + m_blk + tid];
    smm[tid] = sum_t[b * N_ + m_blk + tid];
    idm[tid] = ids_t[b * N_ + m_blk + tid];
  }
  if (tid < BN) {
    sqn[tid] = sq_t1 [b * N_ + n_blk + tid];
    smn[tid] = sum_t1[b * N_ + n_blk + tid];
    idn[tid] = ids_t1[b * N_ + n_blk + tid];
  }

  // 8 waves: 4 row-groups x 2 col-groups, each wave owns a 32x32 output tile.
  const int m0 = (wave & 3) * 32;
  const int n0 = (wave >> 2) * 32;

  v8f c00 = {}; v8f c01 = {}; v8f c10 = {}; v8f c11 = {};

#pragma unroll
  for (int p = 0; p < NPANEL; ++p) {
    const int cur = p & 1;
    if (p + 1 < NPANEL) {
      issue_panel((p + 1) * BK, cur ^ 1);   // stream next panel into other buffer
      wait_asynccnt_6();                    // in-order completion => panel p landed
    } else {
      wait_asynccnt_0();
    }
    __syncthreads();

    const float* as = As[cur];
    const float* bs = Bs[cur];
#pragma unroll
    for (int k0 = 0; k0 < BK; k0 += 4) {
      const int ka = k0 + 2 * hlf;
      const v2f a0 = { as[(m0      + l16) * LDP + ka], as[(m0      + l16) * LDP + ka + 1] };
      const v2f a1 = { as[(m0 + 16 + l16) * LDP + ka], as[(m0 + 16 + l16) * LDP + ka + 1] };
      const v2f b0 = { bs[(n0      + l16) * LDP + ka], bs[(n0      + l16) * LDP + ka + 1] };
      const v2f b1 = { bs[(n0 + 16 + l16) * LDP + ka], bs[(n0 + 16 + l16) * LDP + ka + 1] };
      c00 = __builtin_amdgcn_wmma_f32_16x16x4_f32(false, a0, false, b0, (short)0, c00, false, false);
      c01 = __builtin_amdgcn_wmma_f32_16x16x4_f32(false, a0, false, b1, (short)0, c01, false, false);
      c10 = __builtin_amdgcn_wmma_f32_16x16x4_f32(false, a1, false, b0, (short)0, c10, false, false);
      c11 = __builtin_amdgcn_wmma_f32_16x16x4_f32(false, a1, false, b1, (short)0, c11, false, false);
    }
    __syncthreads();   // all waves done reading buf[cur] before it is re-filled
  }

  // Fused epilogue: d2 -> dist -> hinge/positive loss, masked by IDs.
  float lsum = 0.f;
  const float cEps = (float)D_ * EPS_ * EPS_;
#pragma unroll
  for (int t = 0; t < 4; ++t) {
    const v8f& c = (t == 0) ? c00 : (t == 1) ? c01 : (t == 2) ? c10 : c11;
    const int Mt = m0 + ((t >> 1) ? 16 : 0);
    const int Nt = n0 + ((t & 1) ? 16 : 0);
    const int nc = Nt + l16;
    const float sq_n = sqn[nc], sm_n = smn[nc];
    const int   id_n = idn[nc];
#pragma unroll
    for (int r = 0; r < 8; ++r) {
      const int   mr  = Mt + hlf * 8 + r;
      const float dot = c[r];
      float d2 = sqm[mr] + sq_n - 2.f * dot + 2.f * EPS_ * (smm[mr] - sm_n) + cEps;
      d2 = fmaxf(d2, 0.f);
      const float dist  = sqrtf(d2);
      const float hinge = fmaxf(MARGIN_ - dist, 0.f);
      const int   id_m  = idm[mr];
      const float lm = (id_m == id_n) ? d2 : hinge * hinge;   // dist^2 == clamped d2
      lsum += ((id_m > 0) && (id_n > 0)) ? lm : 0.f;
    }
  }

  // Deterministic block reduction.
  red[tid] = lsum;
  __syncthreads();
  for (int s = 128; s > 0; s >>= 1) {
    if (tid < s) red[tid] += red[tid + s];
    __syncthreads();
  }
  if (tid == 0) {
    const int blk = (blockIdx.z * gridDim.y + blockIdx.y) * gridDim.x + blockIdx.x;
    partials[blk] = red[0];
  }
}

// ---------------------------------------------------------------------------
// Kernel 3: fixed-order reduction of partials + pair count -> scalar loss.
// ---------------------------------------------------------------------------
__global__ void __launch_bounds__(256)
finalize_kernel(const float* __restrict__ partials, int nPartials,
                const int* __restrict__ ids_t, const int* __restrict__ ids_t1,
                float* __restrict__ out) {
  __shared__ float red[256];
  __shared__ int cnts[2 * B_];
  const int tid = threadIdx.x;
  if (tid < 2 * B_) cnts[tid] = 0;

  float s = 0.f;
  for (int i = tid; i < nPartials; i += 256) s += partials[i];  // fixed order
  red[tid] = s;
  __syncthreads();

  // Per-batch valid-ID counts (integer atomics: order-independent).
  int lc_t[B_];  int lc_t1[B_];
#pragma unroll
  for (int b = 0; b < B_; ++b) { lc_t[b] = 0; lc_t1[b] = 0; }
  for (int i = tid; i < B_ * N_; i += 256) {
    const int b = i >> 11;                 // N_ = 2048
    lc_t[b]  += (ids_t[i]  > 0) ? 1 : 0;
    lc_t1[b] += (ids_t1[i] > 0) ? 1 : 0;
  }
#pragma unroll
  for (int b = 0; b < B_; ++b) {
    atomicAdd(&cnts[b], lc_t[b]);
    atomicAdd(&cnts[B_ + b], lc_t1[b]);
  }
  __syncthreads();

  for (int sft = 128; sft > 0; sft >>= 1) {
    if (tid < sft) red[tid] += red[tid + sft];
    __syncthreads();
  }
  if (tid == 0) {
    double np = 0.0;
    for (int b = 0; b < B_; ++b) np += (double)cnts[b] * (double)cnts[B_ + b];
    out[0] = (np == 0.0) ? 0.f : (float)((double)red[0] / np);
  }
}

// ---------------------------------------------------------------------------
extern "C" void kernel_launch(void* const* d_in, const int* in_sizes, int n_in,
                              void* d_out, int out_size, void* d_ws, size_t ws_size,
                              hipStream_t stream) {
  const float* et     = (const float*)d_in[0];
  const float* et1    = (const float*)d_in[1];
  const int*   ids_t  = (const int*)d_in[2];
  const int*   ids_t1 = (const int*)d_in[3];

  float* ws      = (float*)d_ws;
  float* sq_t    = ws;
  float* sum_t   = sq_t   + B_ * N_;
  float* sq_t1   = sum_t  + B_ * N_;
  float* sum_t1  = sq_t1  + B_ * N_;
  float* partials = sum_t1 + B_ * N_;

  const int nStatBlocks = (2 * B_ * N_) / 8;                 // 4096
  rowstats_kernel<<<nStatBlocks, 256, 0, stream>>>(et, et1, sq_t, sum_t, sq_t1, sum_t1);

  dim3 grid(N_ / BM, N_ / BN, B_);                           // 16 x 32 x 8
  loss_gemm_kernel<<<grid, 256, 0, stream>>>(et, et1, ids_t, ids_t1,
                                             sq_t, sum_t, sq_t1, sum_t1, partials);

  const int nPartials = (N_ / BM) * (N_ / BN) * B_;          // 4096
  finalize_kernel<<<1, 256, 0, stream>>>(partials, nPartials, ids_t, ids_t1, (float*)d_out);
}